// I2DEtAl_46188078301832
// MI455X (gfx1250) — compile-verified
//
#include <hip/hip_runtime.h>
#include <hip/hip_bf16.h>

typedef __attribute__((ext_vector_type(2))) float v2f;
typedef __attribute__((ext_vector_type(8))) float v8f;

#define WAVES_PER_BLOCK 8

// ---------------------------------------------------------------------------
// Batched fp32 WMMA GEMM:  Y[b] = W[MxK] @ X[b][KxN]   (all dims compile-time)
//   grid.x = B, 256 threads (8 waves).
//   Wave tiling: each wave owns ONE 16-row m-tile (per m-iteration) and NT
//   n-tiles. The A fragment is loaded once per k-step and reused by NT
//   back-to-back v_wmma on independent accumulators -> deep WMMA interleave,
//   immediate-offset B loads (N is constexpr).
//   V_WMMA_F32_16X16X4_F32 fragment layouts per CDNA5 ISA 7.12.2:
//     A 16x4 : lanes 0-15 -> M rows, vgpr0/1 = K{0,1}; lanes 16-31 = K{2,3}
//     B 4x16 : lanes 0-15 -> N cols with K{0,1}; lanes 16-31 -> K{2,3}
//     C 16x16: vgpr j -> M = j (lanes 0-15) / 8+j (lanes 16-31), N = lane%16
// ---------------------------------------------------------------------------
template <int M, int N, int K>
__global__ __launch_bounds__(256) void inv_gemm_wmma(
    const float* __restrict__ W,   // [M,K] row-major
    const float* __restrict__ X,   // [B,K,N]
    float* __restrict__ Y)         // [B,M,N]
{
    constexpr int TM  = M / 16;                       // m tiles
    constexpr int TN  = N / 16;                       // n tiles
    constexpr int WPC = (TM < 8) ? TM : 8;            // waves along M
    constexpr int GRP = 8 / WPC;                      // wave groups along N
    constexpr int NT  = TN / GRP;                     // n tiles per wave
    constexpr int MI  = TM / WPC;                     // m iterations per wave
    static_assert(GRP * NT == TN && WPC * MI == TM, "bad tiling");

    const int b    = blockIdx.x;
    const int lane = threadIdx.x & 31;
    const int wave = threadIdx.x >> 5;
    const int half = lane >> 4;
    const int l16  = lane & 15;

    const float* __restrict__ Xb = X + (size_t)b * K * N;
    float*       __restrict__ Yb = Y + (size_t)b * M * N;

    const int mwave = wave % WPC;
    const int group = wave / WPC;
    const int ncol0 = group * (NT * 16);

#pragma unroll
    for (int mi = 0; mi < MI; ++mi) {
        const int tm  = mi * WPC + mwave;
        const float* __restrict__ Wr = W + (size_t)(tm * 16 + l16) * K;

        v8f acc[NT] = {};

#pragma unroll 4
        for (int k0 = 0; k0 < K; k0 += 4) {
            const int ka = k0 + (half << 1);
            v2f a;
            a.x = Wr[ka];
            a.y = Wr[ka + 1];
            const float* __restrict__ Xk = Xb + (size_t)ka * N + ncol0 + l16;
            if ((k0 & 31) == 0)   // stream the activation panel ahead
                __builtin_prefetch(Xb + (size_t)(k0 + 32) * N + ncol0 + l16, 0, 1);
#pragma unroll
            for (int nt = 0; nt < NT; ++nt) {
                v2f bf;
                bf.x = Xk[nt * 16];
                bf.y = Xk[nt * 16 + N];
                acc[nt] = __builtin_amdgcn_wmma_f32_16x16x4_f32(
                              false, a, false, bf, (short)0, acc[nt],
                              false, false);
            }
        }

        const int orow = tm * 16 + (half << 3);
#pragma unroll
        for (int nt = 0; nt < NT; ++nt) {
#pragma unroll
            for (int j = 0; j < 8; ++j)
                Yb[(size_t)(orow + j) * N + ncol0 + nt * 16 + l16] = acc[nt][j];
        }
    }
}

// ---------------------------------------------------------------------------
// 2x2 average pool (stride 2):  X[B,C,H,W] -> Y[B,C,H/2,W/2]
// ---------------------------------------------------------------------------
template <int H, int W>
__global__ void avgpool2x2(const float* __restrict__ X, float* __restrict__ Y,
                           int total)
{
    int i = blockIdx.x * blockDim.x + threadIdx.x;
    if (i >= total) return;
    constexpr int Wo = W >> 1, Ho = H >> 1;
    int wo = i % Wo; int t = i / Wo;
    int ho = t % Ho; t /= Ho;          // t = b*C + c
    const float* p = X + ((size_t)t * H + (ho << 1)) * W + (wo << 1);
    Y[i] = 0.25f * (p[0] + p[1] + p[W] + p[W + 1]);
}

// ---------------------------------------------------------------------------
// BatchNorm batch statistics over (B, Ho, Wo) per channel -> scale/shift
//   scale = gamma * rsqrt(var + eps); shift = beta - mean*scale
// ---------------------------------------------------------------------------
template <int C, int HW, int B>
__global__ __launch_bounds__(256) void bn_stats(
    const float* __restrict__ R, const float* __restrict__ gamma,
    const float* __restrict__ beta, float* __restrict__ scale,
    float* __restrict__ shift)
{
    __shared__ float s_sum[256];
    __shared__ float s_sq[256];
    const int c = blockIdx.x;
    const int t = threadIdx.x;
    constexpr int n = B * HW;
    float sum = 0.f, sq = 0.f;
    for (int i = t; i < n; i += 256) {
        const int b = i / HW, p = i % HW;      // HW is a power of two -> shifts
        const float v = R[((size_t)b * C + c) * HW + p];
        sum += v; sq += v * v;
    }
    s_sum[t] = sum; s_sq[t] = sq;
    __syncthreads();
    for (int off = 128; off > 0; off >>= 1) {
        if (t < off) { s_sum[t] += s_sum[t + off]; s_sq[t] += s_sq[t + off]; }
        __syncthreads();
    }
    if (t == 0) {
        constexpr float inv = 1.0f / (float)n;
        const float mean = s_sum[0] * inv;
        const float var  = s_sq[0] * inv - mean * mean;
        const float sc   = gamma[c] * rsqrtf(var + 1e-5f);
        scale[c] = sc;
        shift[c] = beta[c] - mean * sc;
    }
}

// ---------------------------------------------------------------------------
// Involution apply: per (b, ho, wo)
//   act[o]   = relu(scale[o]*r[b,o,ho,wo] + shift[o])
//   kern[kk] = sum_o Wspan[kk,o] * act[o]
//   out[b,o] = sum_{ki,kj} kern[ki*k+kj] * xi[b,o, ho*s-p+ki, wo*s-p+kj]
// ---------------------------------------------------------------------------
template <int C, int Hin, int Ho, int KS, int S, int P>
__global__ __launch_bounds__(256) void inv_apply(
    const float* __restrict__ XI,    // [B,C,Hin,Hin]
    const float* __restrict__ R,     // [B,C,Ho,Ho]
    const float* __restrict__ scale, const float* __restrict__ shift,
    const float* __restrict__ Wspan, // [KS*KS, C]
    float* __restrict__ OUT)         // [B,C,Ho,Ho]
{
    __shared__ float s_act[256];
    __shared__ float s_kern[KS * KS];
    const int b  = blockIdx.x;
    const int hw = blockIdx.y;
    const int h  = hw / Ho, w = hw % Ho;
    const int t  = threadIdx.x;
    constexpr int HWo = Ho * Ho;

    if (t < C) {
        float v = scale[t] * R[((size_t)b * C + t) * HWo + hw] + shift[t];
        s_act[t] = v > 0.f ? v : 0.f;
    }
    __syncthreads();

    if (t < KS * KS) {
        float acc = 0.f;
        for (int o = 0; o < C; ++o) acc += Wspan[t * C + o] * s_act[o];
        s_kern[t] = acc;
    }
    __syncthreads();

    for (int o = t; o < C; o += 256) {
        float acc = 0.f;
#pragma unroll
        for (int ki = 0; ki < KS; ++ki) {
            const int yi = h * S - P + ki;
            if (yi < 0 || yi >= Hin) continue;
#pragma unroll
            for (int kj = 0; kj < KS; ++kj) {
                const int xj = w * S - P + kj;
                if (xj < 0 || xj >= Hin) continue;
                acc += s_kern[ki * KS + kj] *
                       XI[(((size_t)b * C + o) * Hin + yi) * Hin + xj];
            }
        }
        OUT[((size_t)b * C + o) * HWo + hw] = acc;
    }
}

// ---------------------------------------------------------------------------
// FC via WMMA: OUT[b,cls] = sum_f H[b,f]*Wfc[cls,f] + bias[cls]
//   M = 16 classes (1 tile), N = batch, K = F = 4096 (constexpr).
// ---------------------------------------------------------------------------
template <int F>
__global__ __launch_bounds__(256) void fc_wmma(
    const float* __restrict__ Hm,   // [B, F]
    const float* __restrict__ Wfc,  // [16, F]
    const float* __restrict__ bias, // [16]
    float* __restrict__ OUT,        // [B, 16]
    int Bn)
{
    const int lane = threadIdx.x & 31;
    const int wave = threadIdx.x >> 5;
    const int half = lane >> 4, l16 = lane & 15;
    const int n0 = (blockIdx.x * WAVES_PER_BLOCK + wave) << 4;
    if (n0 >= Bn) return;

    const float* __restrict__ Wr = Wfc + (size_t)l16 * F;        // A row
    const float* __restrict__ Hr = Hm + (size_t)(n0 + l16) * F;  // B col (h^T)

    v8f c = {};
#pragma unroll 4
    for (int k0 = 0; k0 < F; k0 += 4) {
        const int ka = k0 + (half << 1);
        v2f a, bf;
        a.x  = Wr[ka];  a.y  = Wr[ka + 1];
        bf.x = Hr[ka];  bf.y = Hr[ka + 1];
        c = __builtin_amdgcn_wmma_f32_16x16x4_f32(
                false, a, false, bf, (short)0, c, false, false);
    }
#pragma unroll
    for (int j = 0; j < 8; ++j) {
        const int cls = (half << 3) + j;
        OUT[(size_t)(n0 + l16) * 16 + cls] = c[j] + bias[cls];
    }
}

// ---------------------------------------------------------------------------
// Host orchestration
// ---------------------------------------------------------------------------
extern "C" void kernel_launch(void* const* d_in, const int* in_sizes, int n_in,
                              void* d_out, int out_size, void* d_ws, size_t ws_size,
                              hipStream_t stream)
{
    const float* x   = (const float*)d_in[0];
    const float* Wi1 = (const float*)d_in[1];
    const float* Wr1 = (const float*)d_in[2];
    const float* g1  = (const float*)d_in[3];
    const float* be1 = (const float*)d_in[4];
    const float* Ws1 = (const float*)d_in[5];
    const float* Wi2 = (const float*)d_in[6];
    const float* Wr2 = (const float*)d_in[7];
    const float* g2  = (const float*)d_in[8];
    const float* be2 = (const float*)d_in[9];
    const float* Ws2 = (const float*)d_in[10];
    const float* Wi3 = (const float*)d_in[11];
    const float* Wr3 = (const float*)d_in[12];
    const float* g3  = (const float*)d_in[13];
    const float* be3 = (const float*)d_in[14];
    const float* Ws3 = (const float*)d_in[15];
    const float* Wfc = (const float*)d_in[16];
    const float* bfc = (const float*)d_in[17];
    float* out = (float*)d_out;

    const int B = 1024;

    // Workspace arenas (floats). Peak ~161 MB.
    float* ws    = (float*)d_ws;
    float* xi    = ws;                               // <= 1024*64*256  = 16.78M
    float* xp    = xi + (size_t)1024 * 64 * 256;     // <= 1024*200*64  = 13.11M
    float* rr    = xp + (size_t)1024 * 200 * 64;     // <= 1024*64*64   =  4.19M
    float* o1    = rr + (size_t)1024 * 64 * 64;      //    1024*64*64   =  4.19M
    float* o2    = o1 + (size_t)1024 * 64 * 64;      //    1024*128*16  =  2.10M
    float* stats = o2 + (size_t)1024 * 128 * 16;     //    1K
    float* o3    = xp;                               // overlay: xp dead in layer 3
    float* scale = stats;
    float* shift = stats + 256;

    // ---------------- Layer 1: Cin=200 Cout=64 H=16 k=2 s=2 p=0 -> 8x8 ------
    inv_gemm_wmma<64, 256, 200><<<B, 256, 0, stream>>>(Wi1, x, xi);
    {
        const int total = 1024 * 200 * 8 * 8;
        avgpool2x2<16, 16><<<(total + 255) / 256, 256, 0, stream>>>(x, xp, total);
    }
    inv_gemm_wmma<64, 64, 200><<<B, 256, 0, stream>>>(Wr1, xp, rr);
    bn_stats<64, 64, 1024><<<64, 256, 0, stream>>>(rr, g1, be1, scale, shift);
    inv_apply<64, 16, 8, 2, 2, 0><<<dim3(B, 64), 256, 0, stream>>>(
        xi, rr, scale, shift, Ws1, o1);

    // ---------------- Layer 2: Cin=64 Cout=128 H=8 k=2 s=2 p=0 -> 4x4 -------
    inv_gemm_wmma<128, 64, 64><<<B, 256, 0, stream>>>(Wi2, o1, xi);
    {
        const int total = 1024 * 64 * 4 * 4;
        avgpool2x2<8, 8><<<(total + 255) / 256, 256, 0, stream>>>(o1, xp, total);
    }
    inv_gemm_wmma<128, 16, 64><<<B, 256, 0, stream>>>(Wr2, xp, rr);
    bn_stats<128, 16, 1024><<<128, 256, 0, stream>>>(rr, g2, be2, scale, shift);
    inv_apply<128, 8, 4, 2, 2, 0><<<dim3(B, 16), 256, 0, stream>>>(
        xi, rr, scale, shift, Ws2, o2);

    // ---------------- Layer 3: Cin=128 Cout=256 H=4 k=3 s=1 p=1 -> 4x4 ------
    inv_gemm_wmma<256, 16, 128><<<B, 256, 0, stream>>>(Wi3, o2, xi);
    inv_gemm_wmma<256, 16, 128><<<B, 256, 0, stream>>>(Wr3, o2, rr);  // s=1: xp=x
    bn_stats<256, 16, 1024><<<256, 256, 0, stream>>>(rr, g3, be3, scale, shift);
    inv_apply<256, 4, 4, 3, 1, 1><<<dim3(B, 16), 256, 0, stream>>>(
        xi, rr, scale, shift, Ws3, o3);

    // ---------------- FC: [1024,4096] @ [4096,16]^T + b --------------------
    fc_wmma<256 * 4 * 4><<<B / 16 / WAVES_PER_BLOCK, 256, 0, stream>>>(
        o3, Wfc, bfc, out, B);
}